// AutoCorrelation_18580028522947
// MI455X (gfx1250) — compile-verified
//
#include <hip/hip_runtime.h>
#include <hip/hip_bf16.h>
#include <stdint.h>

// Problem constants (from reference)
#define BATCH   16
#define SEQ     1024
#define DMODEL  512
#define NHEAD   8
#define DHEAD   64          // DMODEL / NHEAD
#define NBH     (BATCH * NHEAD)   // 128
#define TOPK    13          // int(2 * ln(1024)) = 13

// corr kernel tiling
#define U_CHUNK  256        // k rows staged per pass (4 passes)
#define TAU_BLK  64         // lags per tile; 16 lags/thread-group, 4 groups
#define QWIN     320        // q window rows: U_CHUNK + TAU_BLK - 1 = 319 -> pad 320

// ---------------------------------------------------------------------------
// CDNA5 async global->LDS helpers (gfx1250). VDST VGPR carries the
// wave-relative LDS byte offset (low 32 bits of the generic shared address);
// GV addressing: 64-bit global VA in a VGPR pair, saddr = off.
// Tracked by ASYNCcnt; drained with s_wait_asynccnt.
// ---------------------------------------------------------------------------
__device__ __forceinline__ void async_copy_b128(void* lds_dst, const void* gsrc) {
    unsigned lds_off = (unsigned)(uintptr_t)lds_dst;   // low 32 bits = LDS offset
    asm volatile("global_load_async_to_lds_b128 %0, %1, off"
                 :: "v"(lds_off), "v"(gsrc)
                 : "memory");
}

__device__ __forceinline__ void wait_async0() {
    asm volatile("s_wait_asynccnt 0x0" ::: "memory");
}

__device__ __forceinline__ float2 ld2(const float* p) {
    return *(const float2*)p;
}

// ---------------------------------------------------------------------------
// Kernel 1: circular correlation  corr[bh, tau, d] = sum_u q[(u+tau)%L, d]*k[u, d]
// Grid (bh, tau-quarter); 128 threads = 4 waves. Each lane owns channel pair
// (2l, 2l+1): LDS traffic is ds_load_b64, conflict-free (32 lanes x 2 banks),
// FMAs are float2 (dual-issue / packed on gfx1250).
// K staged in 256-row chunks and Q in 320-row sliding windows via async-to-LDS.
// ---------------------------------------------------------------------------
__global__ void __launch_bounds__(128)
corr_kernel(const float* __restrict__ Q, const float* __restrict__ K,
            float* __restrict__ corr) {
    __shared__ float kbuf[U_CHUNK * DHEAD];   //  64 KB
    __shared__ float qbuf[QWIN   * DHEAD];    //  80 KB  (total 144 KB)

    const int bh   = blockIdx.x;
    const int b    = bh >> 3;
    const int h    = bh & 7;
    const int tid  = threadIdx.x;
    const int lane = tid & 31;
    const int d0   = lane * 2;         // channel pair owned by this lane
    const int g    = tid >> 5;         // lag-group 0..3 (16 lags each)
    const int qb   = g * 16;           // tau offset base within tile
    const int tb0  = blockIdx.y * 4;   // this block's 4 lag tiles (of 16 total)

    const float* Qp = Q + (size_t)b * SEQ * DMODEL + h * DHEAD;
    const float* Kp = K + (size_t)b * SEQ * DMODEL + h * DHEAD;
    float*       Cp = corr + (size_t)bh * SEQ * DHEAD;

    for (int p = 0; p < 4; ++p) {                    // k chunk: rows [p*256, p*256+256)
        // ---- stage K chunk: 256 rows * 16 float4 = 4096 float4, 32 per thread
        for (int it = 0; it < 32; ++it) {
            int j   = tid + it * 128;
            int row = j >> 4;
            int c4  = j & 15;
            async_copy_b128(&kbuf[j * 4], Kp + (size_t)(p * U_CHUNK + row) * DMODEL + c4 * 4);
        }

        for (int tbl = 0; tbl < 4; ++tbl) {
            const int tb = tb0 + tbl;
            // ---- stage Q window: rows s = (p*256 + tb*64 + w) mod 1024, w < 320
            const int qbase = (p * U_CHUNK + tb * TAU_BLK) & (SEQ - 1);
            for (int it = 0; it < 40; ++it) {
                int j  = tid + it * 128;
                int w  = j >> 4;
                int c4 = j & 15;
                int s  = (qbase + w) & (SEQ - 1);
                async_copy_b128(&qbuf[j * 4], Qp + (size_t)s * DMODEL + c4 * 4);
            }
            wait_async0();           // our async writes (k chunk on tbl==0, q window)
            __syncthreads();         // everyone's writes visible

            // ---- compute: acc[r] = sum_u k[u]*q[u + qb + r], register ring of 16 q
            float2 acc[16];
            #pragma unroll
            for (int r = 0; r < 16; ++r) { acc[r].x = 0.0f; acc[r].y = 0.0f; }

            float2 qr[16];
            #pragma unroll
            for (int j = 0; j < 16; ++j) qr[j] = ld2(&qbuf[(qb + j) * DHEAD + d0]);

            for (int uo = 0; uo < 16; ++uo) {
                #pragma unroll
                for (int ui = 0; ui < 16; ++ui) {
                    const int u = uo * 16 + ui;
                    const float2 kv = ld2(&kbuf[u * DHEAD + d0]);
                    #pragma unroll
                    for (int r = 0; r < 16; ++r) {
                        const float2 qv = qr[(ui + r) & 15];
                        acc[r].x = fmaf(kv.x, qv.x, acc[r].x);
                        acc[r].y = fmaf(kv.y, qv.y, acc[r].y);
                    }
                    qr[ui] = ld2(&qbuf[(u + 16 + qb) * DHEAD + d0]);  // max row 319
                }
            }

            // ---- accumulate into corr workspace (same thread owns (tau,d) each pass)
            const int tau0 = tb * TAU_BLK + qb;
            if (p == 0) {
                #pragma unroll
                for (int r = 0; r < 16; ++r)
                    *(float2*)&Cp[(size_t)(tau0 + r) * DHEAD + d0] = acc[r];
            } else {
                #pragma unroll
                for (int r = 0; r < 16; ++r) {
                    float2* o = (float2*)&Cp[(size_t)(tau0 + r) * DHEAD + d0];
                    float2 c = *o;
                    c.x += acc[r].x; c.y += acc[r].y;
                    *o = c;
                }
            }
            __syncthreads();         // safe to overwrite qbuf (and kbuf on last tbl)
        }
    }
}

// ---------------------------------------------------------------------------
// Kernel 2: per-(b,h,d) top-13 over lags + softmax.
// One lane per series; lane = d so the 1024 strided reads are coalesced.
// ---------------------------------------------------------------------------
__global__ void __launch_bounds__(64)
topk_kernel(const float* __restrict__ corr,
            float* __restrict__ wout, int* __restrict__ iout) {
    const int bh = blockIdx.x;
    const int d  = threadIdx.x;      // 0..63
    const float* Cp = corr + (size_t)bh * SEQ * DHEAD + d;

    float tv[TOPK];
    int   ti[TOPK];
    #pragma unroll
    for (int j = 0; j < TOPK; ++j) { tv[j] = -3.4e38f; ti[j] = 0; }

    for (int t = 0; t < SEQ; ++t) {
        float nv = Cp[(size_t)t * DHEAD];
        int   nt = t;
        #pragma unroll
        for (int j = 0; j < TOPK; ++j) {
            if (nv > tv[j]) {
                float fv = tv[j]; int fi = ti[j];
                tv[j] = nv; ti[j] = nt;
                nv = fv; nt = fi;
            }
        }
    }

    // softmax over the 13 (tv[0] is the max: list kept sorted descending)
    const float m = tv[0];
    float s = 0.0f;
    float e[TOPK];
    #pragma unroll
    for (int j = 0; j < TOPK; ++j) { e[j] = __expf(tv[j] - m); s += e[j]; }
    const float inv = 1.0f / s;

    float* Wp = wout + (size_t)bh * TOPK * DHEAD;
    int*   Ip = iout + (size_t)bh * TOPK * DHEAD;
    #pragma unroll
    for (int j = 0; j < TOPK; ++j) {
        Wp[j * DHEAD + d] = e[j] * inv;
        Ip[j * DHEAD + d] = ti[j];
    }
}

// ---------------------------------------------------------------------------
// Kernel 3: time-delay aggregation.
// out[b, t, h*64+d] = sum_i w[i,d] * V[b, min(idx[i,d]+t, L-1), h*64+d]
// Per-lane gather rows differ, but V (100 MB of inputs total) is L2-resident
// (192 MB global L2), so the 13x gather re-reads stay on-chip.
// ---------------------------------------------------------------------------
__global__ void __launch_bounds__(256)
agg_kernel(const float* __restrict__ V,
           const float* __restrict__ win, const int* __restrict__ iin,
           float* __restrict__ out) {
    const int bh  = blockIdx.x;
    const int b   = bh >> 3;
    const int h   = bh & 7;
    const int tid = threadIdx.x;
    const int d   = tid & 63;
    const int g   = tid >> 6;

    const float* Vp = V   + (size_t)b * SEQ * DMODEL + h * DHEAD + d;
    float*       Op = out + (size_t)b * SEQ * DMODEL + h * DHEAD + d;

    float wr[TOPK];
    int   ir[TOPK];
    const float* Wp = win + (size_t)bh * TOPK * DHEAD + d;
    const int*   Ip = iin + (size_t)bh * TOPK * DHEAD + d;
    #pragma unroll
    for (int j = 0; j < TOPK; ++j) {
        wr[j] = Wp[j * DHEAD];
        ir[j] = Ip[j * DHEAD];
    }

    const int t0 = blockIdx.y * 128 + g * 32;
    for (int i = 0; i < 32; ++i) {
        const int t = t0 + i;
        float acc = 0.0f;
        #pragma unroll
        for (int j = 0; j < TOPK; ++j) {
            int row = ir[j] + t;
            row = (row > SEQ - 1) ? (SEQ - 1) : row;
            acc = fmaf(wr[j], Vp[(size_t)row * DMODEL], acc);
        }
        Op[(size_t)t * DMODEL] = acc;
    }
}

// ---------------------------------------------------------------------------
extern "C" void kernel_launch(void* const* d_in, const int* in_sizes, int n_in,
                              void* d_out, int out_size, void* d_ws, size_t ws_size,
                              hipStream_t stream) {
    (void)in_sizes; (void)n_in; (void)out_size; (void)ws_size;

    const float* Q = (const float*)d_in[0];
    const float* K = (const float*)d_in[1];
    const float* V = (const float*)d_in[2];
    float*       O = (float*)d_out;

    // workspace layout: corr | weights | indices   (~34.4 MB total)
    float* corr = (float*)d_ws;                                  // 128*1024*64 f32
    float* wbuf = corr + (size_t)NBH * SEQ * DHEAD;              // 128*13*64 f32
    int*   ibuf = (int*)(wbuf + (size_t)NBH * TOPK * DHEAD);     // 128*13*64 i32

    corr_kernel<<<dim3(NBH, 4), 128, 0, stream>>>(Q, K, corr);
    topk_kernel<<<NBH, 64, 0, stream>>>(corr, wbuf, ibuf);
    agg_kernel<<<dim3(NBH, 8), 256, 0, stream>>>(V, wbuf, ibuf, O);
}